// EX_Module1_7670811590905
// MI455X (gfx1250) — compile-verified
//
#include <hip/hip_runtime.h>
#include <math.h>

typedef float v2f __attribute__((ext_vector_type(2)));
typedef float v8f __attribute__((ext_vector_type(8)));
typedef int   v2i __attribute__((ext_vector_type(2)));
typedef int   v4i __attribute__((ext_vector_type(4)));

#define C 512
#define HW 4096
#define NB 8

#ifndef __has_builtin
#define __has_builtin(x) 0
#endif

#if __has_builtin(__builtin_amdgcn_global_load_async_to_lds_b128) && \
    __has_builtin(__builtin_amdgcn_global_load_async_to_lds_b64)
#define HAVE_ASYNC_LDS 1
#else
#define HAVE_ASYNC_LDS 0
#endif

#if HAVE_ASYNC_LDS
#if __has_builtin(__builtin_amdgcn_s_wait_asynccnt)
#define WAIT_ASYNC() __builtin_amdgcn_s_wait_asynccnt(0)
#else
#define WAIT_ASYNC() asm volatile("s_wait_asynccnt 0x0" ::: "memory")
#endif
#else
#define WAIT_ASYNC()
#endif

__device__ __forceinline__ float hsig(float v) {
    return fminf(fmaxf((v + 3.0f) * (1.0f / 6.0f), 0.0f), 1.0f);
}

// 256-thread block sum reduction; safe for back-to-back reuse of sm.
__device__ __forceinline__ float block_reduce_sum(float v, float* sm) {
    int t = threadIdx.x;
    sm[t] = v;
    __syncthreads();
    for (int s = 128; s > 0; s >>= 1) {
        if (t < s) sm[t] += sm[t + s];
        __syncthreads();
    }
    float r = sm[0];
    __syncthreads();
    return r;
}

// ---------------- Kernel A: g[b,c] = mean over h*w ----------------
__global__ __launch_bounds__(256) void k_gmean(const float* __restrict__ x,
                                               float* __restrict__ g) {
    int blk = blockIdx.x;                    // b*512 + c
    const float* p = x + (size_t)blk * HW;
    float s = 0.f;
    for (int i = threadIdx.x; i < HW; i += 256) s += p[i];
    __shared__ float sm[256];
    s = block_reduce_sum(s, sm);
    if (threadIdx.x == 0) g[blk] = s * (1.0f / HW);
}

// ---------------- Kernel B: spat[b,p] = hsig(sum_c x[b,c,p]*wc[c]) ----------------
__global__ __launch_bounds__(256) void k_spatial(const float* __restrict__ x,
                                                 const float* __restrict__ wc,
                                                 float* __restrict__ spat) {
    int b = blockIdx.y;
    int p0 = blockIdx.x * 512;
    int t = threadIdx.x;
    const float* xb = x + (size_t)b * C * HW + p0;
    float a0 = 0.f, a1 = 0.f;
    for (int c = 0; c < C; c++) {
        float w = wc[c];                      // uniform -> scalar load
        const float* row = xb + (size_t)c * HW;
        a0 = fmaf(row[t], w, a0);
        a1 = fmaf(row[t + 256], w, a1);
    }
    spat[b * HW + p0 + t]       = hsig(a0);
    spat[b * HW + p0 + t + 256] = hsig(a1);
}

// ---------------- Kernel C: channel-attn MLP + LayerNorm ----------------
__global__ __launch_bounds__(256) void k_mlp_ln(const float* __restrict__ g,
                                                const float* __restrict__ w1,
                                                const float* __restrict__ b1,
                                                const float* __restrict__ w2,
                                                const float* __restrict__ b2,
                                                const float* __restrict__ lng,
                                                const float* __restrict__ lnb,
                                                float* __restrict__ ch) {
    int b = blockIdx.x, t = threadIdx.x;
    __shared__ float gs[C];
    __shared__ float h1[C / 2];
    __shared__ float sm[256];
    gs[t] = g[b * C + t];
    gs[t + 256] = g[b * C + t + 256];
    __syncthreads();
    float d = 0.f;
    for (int i = 0; i < C; i++) d = fmaf(gs[i], w1[t * C + i], d);
    h1[t] = fmaxf(d + b1[t], 0.f);
    __syncthreads();
    float d0 = 0.f, d1 = 0.f;
    for (int i = 0; i < C / 2; i++) {
        float v = h1[i];
        d0 = fmaf(v, w2[t * (C / 2) + i], d0);
        d1 = fmaf(v, w2[(t + 256) * (C / 2) + i], d1);
    }
    float c0 = hsig(d0 + b2[t]);
    float c1 = hsig(d1 + b2[t + 256]);
    float s = block_reduce_sum(c0 + c1, sm);
    float mu = s * (1.0f / C);
    float q = block_reduce_sum((c0 - mu) * (c0 - mu) + (c1 - mu) * (c1 - mu), sm);
    float inv = rsqrtf(q * (1.0f / C) + 1e-6f);
    ch[b * C + t]       = (c0 - mu) * inv * lng[t] + lnb[t];
    ch[b * C + t + 256] = (c1 - mu) * inv * lng[t + 256] + lnb[t + 256];
}

// ---------------- Kernel D: per-(b,c) pixel sums of seq/par attention ----------------
__global__ __launch_bounds__(256) void k_parsum(const float* __restrict__ spat,
                                                const float* __restrict__ ch,
                                                float* __restrict__ Sseq,
                                                float* __restrict__ Spar,
                                                float* __restrict__ upre) {
    int c = blockIdx.x, b = blockIdx.y;
    float cv = ch[b * C + c];
    const float* sp = spat + b * HW;
    float ssp = 0.f, spar = 0.f;
    for (int i = threadIdx.x; i < HW; i += 256) {
        float s = sp[i];
        ssp += s;
        spar += floorf((s + cv) * 0.5f);
    }
    __shared__ float sm[256];
    ssp = block_reduce_sum(ssp, sm);
    spar = block_reduce_sum(spar, sm);
    if (threadIdx.x == 0) {
        float sseq = cv * ssp;
        Sseq[b * C + c] = sseq;
        Spar[b * C + c] = spar;
        upre[b * C + c] = (sseq + spar) * (1.0f / HW);
    }
}

// ---------------- Kernel E: SK MLP + softmax + selected-attn channel sums ----------------
__global__ __launch_bounds__(256) void k_sk(const float* __restrict__ upre,
                                            const float* __restrict__ sk1,
                                            const float* __restrict__ sk2,
                                            const float* __restrict__ Sseq,
                                            const float* __restrict__ Spar,
                                            float* __restrict__ aout,
                                            float* __restrict__ selsum) {
    int b = blockIdx.x, t = threadIdx.x;
    __shared__ float u[C];
    __shared__ float h1[C / 2];
    __shared__ float sm[256];
    u[t] = upre[b * C + t];
    u[t + 256] = upre[b * C + t + 256];
    __syncthreads();
    float d = 0.f;
    for (int i = 0; i < C; i++) d = fmaf(u[i], sk1[t * C + i], d);
    h1[t] = fmaxf(d, 0.f);
    __syncthreads();
    float d0 = 0.f, d1 = 0.f;
    for (int i = 0; i < C / 2; i++) {
        float v = h1[i];
        d0 = fmaf(v, sk2[t * (C / 2) + i], d0);
        d1 = fmaf(v, sk2[(t + 256) * (C / 2) + i], d1);
    }
    d0 = fmaxf(d0, 0.f);
    d1 = fmaxf(d1, 0.f);
    // softmax over 512 channels
    sm[t] = fmaxf(d0, d1);
    __syncthreads();
    for (int s = 128; s > 0; s >>= 1) {
        if (t < s) sm[t] = fmaxf(sm[t], sm[t + s]);
        __syncthreads();
    }
    float mx = sm[0];
    __syncthreads();
    float e0 = __expf(d0 - mx), e1 = __expf(d1 - mx);
    float ssum = block_reduce_sum(e0 + e1, sm);
    float a0 = e0 / ssum, a1 = e1 / ssum;
    int i0 = b * C + t, i1 = i0 + 256;
    aout[i0] = a0;
    aout[i1] = a1;
    selsum[i0] = a0 * Sseq[i0] + (1.f - a0) * Spar[i0];
    selsum[i1] = a1 * Sseq[i1] + (1.f - a1) * Spar[i1];
}

// ---------------- Kernel F: WMMA fp32 GEMM (res_w @ x) + fused epilogue ----------------
// out[b,o,p] = x[b,o,p]*sel(b,o,p) + selsum[b,o] + sum_c res_w[o,c]*x[b,c,p]
// Block tile: 32 (o) x 256 (p). 8 wave32s: 2 (o) x 4 (p); wave owns 16x64 via 4 v8f accs.
// Double-buffered LDS staging via GLOBAL_LOAD_ASYNC_TO_LDS (ASYNCcnt) when available.
// B slab LDS layout interleaves row pairs at 4-float blocks so a (K,K+1) fragment is a
// single ds_load_2addr_b32 (offset1 = 4) into one even-aligned VGPR pair:
//   idx(r,p) = (r>>1)*512 + (p>>2)*8 + (r&1)*4 + (p&3)
#define OT 32
#define PT 256
#define KC 16

// Stage one K-chunk: A = res_w[o0..o0+32, kc..kc+16] (row-major 32x16),
//                    B = x[kc..kc+16, p0..p0+256]     (pair-interleaved 16x256).
__device__ __forceinline__ void stage_chunk(const float* __restrict__ xb,
                                            const float* __restrict__ rw,
                                            int o0, int p0, int kc, int t,
                                            float* la, float* lb) {
    // A: 512 floats, 8 bytes per thread, coalesced along k
    int e2 = t * 2;                          // row = e2>>4, col = e2&15 (even)
    const float* ga = &rw[(size_t)(o0 + (e2 >> 4)) * C + kc + (e2 & 15)];
    // B: 4096 floats, 4 x 16 bytes per thread, coalesced along p
    int r4 = (t >> 6) * 4;                   // 0,4,8,12
    int pq = (t & 63) << 2;                  // 0..252 step 4
#if HAVE_ASYNC_LDS
    __builtin_amdgcn_global_load_async_to_lds_b64((v2i*)ga, (v2i*)&la[e2], 0, 0);
#pragma unroll
    for (int rr = 0; rr < 4; rr++) {
        int r = r4 + rr;
        int d = (r >> 1) * (2 * PT) + ((pq >> 2) << 3) + ((r & 1) << 2);
        __builtin_amdgcn_global_load_async_to_lds_b128(
            (v4i*)&xb[(size_t)(kc + r) * HW + p0 + pq],
            (v4i*)&lb[d], 0, 0);
    }
#else
    float2 va = *(const float2*)ga;
    *(float2*)&la[e2] = va;
#pragma unroll
    for (int rr = 0; rr < 4; rr++) {
        int r = r4 + rr;
        int d = (r >> 1) * (2 * PT) + ((pq >> 2) << 3) + ((r & 1) << 2);
        float4 v = *(const float4*)&xb[(size_t)(kc + r) * HW + p0 + pq];
        *(float4*)&lb[d] = v;
    }
#endif
}

__global__ __launch_bounds__(256) void k_gemm_out(const float* __restrict__ x,
                                                  const float* __restrict__ rw,
                                                  const float* __restrict__ spat,
                                                  const float* __restrict__ ch,
                                                  const float* __restrict__ aattn,
                                                  const float* __restrict__ selsum,
                                                  float* __restrict__ out) {
    __shared__ __align__(16) float lds_a[2][OT * KC];   // 2 x 2 KB
    __shared__ __align__(16) float lds_b[2][KC * PT];   // 2 x 16 KB
    int b = blockIdx.z;
    int o0 = blockIdx.y * OT;
    int p0 = blockIdx.x * PT;
    int t = threadIdx.x;
    int lane = t & 31, wave = t >> 5;
    int wo = wave & 1;                       // o sub-tile (16 rows)
    int wp = wave >> 1;                      // p sub-tile (64 cols)

    const float* xb = x + (size_t)b * C * HW;

    v8f acc[4];
    const v8f vzero = {0.f, 0.f, 0.f, 0.f, 0.f, 0.f, 0.f, 0.f};
#pragma unroll
    for (int i = 0; i < 4; i++) acc[i] = vzero;

    int arow = wo * 16 + (lane & 15);
    int hi = lane >> 4;                      // 0 for lanes 0-15, 1 for 16-31
    int pw = wp * 64 + (lane & 15);

    // prologue: fill buffer 0
    stage_chunk(xb, rw, o0, p0, 0, t, lds_a[0], lds_b[0]);

    int buf = 0;
    for (int kc = 0; kc < C; kc += KC, buf ^= 1) {
        WAIT_ASYNC();                        // my async copies for `buf` landed
        __syncthreads();                     // everyone's copies landed
        if (kc + KC < C)                     // overlap next chunk's fill with WMMAs
            stage_chunk(xb, rw, o0, p0, kc + KC, t, lds_a[buf ^ 1], lds_b[buf ^ 1]);

        const float* la = lds_a[buf];
        const float* lb = lds_b[buf];
#pragma unroll
        for (int k4 = 0; k4 < KC; k4 += 4) {
            // A frag: lanes 0-15 -> (K, K+1), lanes 16-31 -> (K+2, K+3)
            v2f afrag = *(const v2f*)&la[arow * KC + k4 + hi * 2];
            int r0 = k4 + hi * 2;            // even
#pragma unroll
            for (int nt = 0; nt < 4; nt++) {
                int pl = pw + nt * 16;
                int bb = (r0 >> 1) * (2 * PT) + ((pl >> 2) << 3) + (pl & 3);
                v2f bfrag;                   // rows r0, r0+1: 16 B apart -> 2addr b32
                bfrag.x = lb[bb];
                bfrag.y = lb[bb + 4];
                acc[nt] = __builtin_amdgcn_wmma_f32_16x16x4_f32(
                    false, afrag, false, bfrag, (short)0, acc[nt], false, false);
            }
        }
        __syncthreads();                     // done reading `buf` before next overwrite
    }

    // Fused epilogue. C/D layout: VGPR r, lane l -> M = r + 8*(l>=16), N = l%16.
    int obase = o0 + wo * 16 + (hi << 3);
    int pbase = p0 + wp * 64;
    float cv8[8], av8[8], ss8[8];
#pragma unroll
    for (int r = 0; r < 8; r++) {
        int o = obase + r;
        cv8[r] = ch[b * C + o];
        av8[r] = aattn[b * C + o];
        ss8[r] = selsum[b * C + o];
    }
#pragma unroll
    for (int nt = 0; nt < 4; nt++) {
        int p = pbase + nt * 16 + (lane & 15);
        float sp = spat[b * HW + p];
#pragma unroll
        for (int r = 0; r < 8; r++) {
            int o = obase + r;
            float seq = cv8[r] * sp;
            float par = floorf((sp + cv8[r]) * 0.5f);
            float sel = av8[r] * seq + (1.f - av8[r]) * par;
            size_t xi = (size_t)b * C * HW + (size_t)o * HW + p;
            out[xi] = x[xi] * sel + ss8[r] + acc[nt][r];
        }
    }
}

extern "C" void kernel_launch(void* const* d_in, const int* in_sizes, int n_in,
                              void* d_out, int out_size, void* d_ws, size_t ws_size,
                              hipStream_t stream) {
    const float* x    = (const float*)d_in[0];
    const float* w_b1 = (const float*)d_in[1];
    const float* b_b1 = (const float*)d_in[2];
    const float* w_b2 = (const float*)d_in[3];
    const float* b_b2 = (const float*)d_in[4];
    const float* wc   = (const float*)d_in[5];
    const float* lng  = (const float*)d_in[6];
    const float* lnb  = (const float*)d_in[7];
    const float* sk1  = (const float*)d_in[8];
    const float* sk2  = (const float*)d_in[9];
    const float* rw   = (const float*)d_in[10];
    float* out = (float*)d_out;

    // Workspace carve-out (floats): 61440 floats = 240 KB total
    float* ws   = (float*)d_ws;
    float* g    = ws;               // 4096
    float* spat = ws + 4096;        // 32768
    float* chv  = ws + 36864;       // 4096
    float* Sseq = ws + 40960;       // 4096
    float* Spar = ws + 45056;       // 4096
    float* upre = ws + 49152;       // 4096
    float* aat  = ws + 53248;       // 4096
    float* ssel = ws + 57344;       // 4096

    k_gmean  <<<dim3(NB * C),        256, 0, stream>>>(x, g);
    k_spatial<<<dim3(8, NB),         256, 0, stream>>>(x, wc, spat);
    k_mlp_ln <<<dim3(NB),            256, 0, stream>>>(g, w_b1, b_b1, w_b2, b_b2, lng, lnb, chv);
    k_parsum <<<dim3(C, NB),         256, 0, stream>>>(spat, chv, Sseq, Spar, upre);
    k_sk     <<<dim3(NB),            256, 0, stream>>>(upre, sk1, sk2, Sseq, Spar, aat, ssel);
    k_gemm_out<<<dim3(HW / PT, C / OT, NB), 256, 0, stream>>>(x, rw, spat, chv, aat, ssel, out);
}